// HexGINLayer_20590073217561
// MI455X (gfx1250) — compile-verified
//
#include <hip/hip_runtime.h>

typedef float v2f __attribute__((ext_vector_type(2)));
typedef float v8f __attribute__((ext_vector_type(8)));

__device__ __forceinline__ v8f wmma4(v2f a, v2f b, v8f c) {
    // D = A(16x4 f32) * B(4x16 f32) + C(16x16 f32)
    return __builtin_amdgcn_wmma_f32_16x16x4_f32(
        /*neg_a=*/false, a, /*neg_b=*/false, b,
        /*c_mod=*/(short)0, c, /*reuse_a=*/false, /*reuse_b=*/false);
}

// ---------------------------------------------------------------------------
// Pack W[K,N] row-major into pair layout Wp[k/2][N][2] so a WMMA B-fragment
// (rows k0,k0+1 at column c) is ONE b64 load with immediate offsets.
__global__ void pack_kernel(const float* __restrict__ W, float* __restrict__ Wp,
                            int K, int N) {
    int idx = blockIdx.x * blockDim.x + threadIdx.x;
    if (idx >= K * N) return;
    int k = idx / N, n = idx - k * N;
    Wp[(size_t)(k >> 1) * (2 * N) + 2 * n + (k & 1)] = W[idx];
}

// ---------------------------------------------------------------------------
__global__ void zero_kernel(float4* __restrict__ p, int n4) {
    int i = blockIdx.x * blockDim.x + threadIdx.x;
    if (i < n4) p[i] = make_float4(0.f, 0.f, 0.f, 0.f);
}

// ---------------------------------------------------------------------------
// Edge scatter: agg[dst] += a_src[src], 128 floats/edge. Coalesced float4
// gather (25.6MB table lives in the 192MB L2) + global_atomic_add_f32.
__global__ void scatter_kernel(const float* __restrict__ a,
                               const int* __restrict__ edges, int E,
                               float* __restrict__ agg) {
    int idx = blockIdx.x * blockDim.x + threadIdx.x;
    if (idx >= E * 32) return;
    int e = idx >> 5;
    int q = (idx & 31) << 2;
    int src = edges[e];
    int dst = edges[E + e];
    const float4 v = *(const float4*)(a + (size_t)src * 128 + q);
    float* d = agg + (size_t)dst * 128 + q;
    atomicAdd(d + 0, v.x);
    atomicAdd(d + 1, v.y);
    atomicAdd(d + 2, v.z);
    atomicAdd(d + 3, v.w);
}

// ---------------------------------------------------------------------------
// C[M,128] = X[M,K] @ W[K,128] + bias.  Wp is pair-packed (2N = 256 pitch).
// 4 waves/block, 16 rows/wave, wave holds all 8 N-tiles (64 acc VGPRs):
// per k-step = 1 A b64 + 8 B b64 + 8 wmma.
template<int K>
__global__ __launch_bounds__(128) void align_kernel(
    const float* __restrict__ X, const float* __restrict__ Wp,
    const float* __restrict__ bias, float* __restrict__ C, int M) {
    const int lane = threadIdx.x & 31;
    const int wid  = threadIdx.x >> 5;
    const int m    = lane & 15;
    const int h    = lane >> 4;
    const int row0 = blockIdx.x * 64 + wid * 16;
    const int rowA = min(row0 + m, M - 1);
    const float* xrow = X + (size_t)rowA * K + 2 * h;
    const float* bp   = Wp + h * 256 + 2 * m;      // (2s+h)*256 + 32t + 2m

    v8f acc[8] = {};
#pragma unroll 4
    for (int s = 0; s < K / 4; ++s) {
        v2f a = *(const v2f*)(xrow + 4 * s);
        const float* bs = bp + (size_t)s * 512;
#pragma unroll
        for (int t = 0; t < 8; ++t) {
            v2f b = *(const v2f*)(bs + 32 * t);
            acc[t] = wmma4(a, b, acc[t]);
        }
    }
#pragma unroll
    for (int t = 0; t < 8; ++t) {
        const int c = t * 16 + m;
        const float bv = bias[c];
#pragma unroll
        for (int j = 0; j < 8; ++j) {
            int r = row0 + j + 8 * h;
            if (r < M) C[(size_t)r * 128 + c] = acc[t][j] + bv;
        }
    }
}

// ---------------------------------------------------------------------------
// Fused GIN MLP:  out = relu([(1+eps)*x | agg] @ W1 + b1) @ W2 + b2
// W1p/W2p pair-packed. 2 waves/block, 16 rows/wave. Hidden bounced through
// LDS (pitch 260 -> conflict-free b64 A-frag reads). Hidden N=256 processed
// as two 128-wide halves (8 acc tiles each).
template<int DSELF>
__global__ __launch_bounds__(64) void mlp_kernel(
    const float* __restrict__ Xd, const float* __restrict__ AGG,
    const float* __restrict__ epsp,
    const float* __restrict__ W1p, const float* __restrict__ b1,
    const float* __restrict__ W2p, const float* __restrict__ b2,
    float* __restrict__ Out, int M) {
    constexpr int K1 = DSELF + 128;
    constexpr int HP = 260;                 // hidden row pitch (floats)
    __shared__ float Hlds[2 * 16 * HP];     // 33.3 KB

    const int lane = threadIdx.x & 31;
    const int wid  = threadIdx.x >> 5;
    const int m    = lane & 15;
    const int h    = lane >> 4;
    const int row0 = blockIdx.x * 32 + wid * 16;
    const int rowA = min(row0 + m, M - 1);
    const float scale = 1.0f + *epsp;

    const float* xrow = Xd  + (size_t)rowA * DSELF + 2 * h;
    const float* grow = AGG + (size_t)rowA * 128 + 2 * h;
    float* hbase = Hlds + wid * 16 * HP;

    // ---- layer 1: hidden[16,256] = relu(concat @ W1 + b1) -> LDS ----
#pragma unroll
    for (int nh = 0; nh < 2; ++nh) {
        const float* bp1 = W1p + h * 512 + 2 * (nh * 128 + m);  // 2N = 512
        v8f acc[8] = {};
#pragma unroll 4
        for (int s = 0; s < DSELF / 4; ++s) {                   // self part
            v2f a = *(const v2f*)(xrow + 4 * s);
            a.x *= scale; a.y *= scale;
            const float* bs = bp1 + (size_t)s * 1024;
#pragma unroll
            for (int t = 0; t < 8; ++t) {
                v2f b = *(const v2f*)(bs + 32 * t);
                acc[t] = wmma4(a, b, acc[t]);
            }
        }
#pragma unroll 4
        for (int s = DSELF / 4; s < K1 / 4; ++s) {              // agg part
            v2f a = *(const v2f*)(grow + 4 * s - DSELF);
            const float* bs = bp1 + (size_t)s * 1024;
#pragma unroll
            for (int t = 0; t < 8; ++t) {
                v2f b = *(const v2f*)(bs + 32 * t);
                acc[t] = wmma4(a, b, acc[t]);
            }
        }
#pragma unroll
        for (int t = 0; t < 8; ++t) {
            const int c = nh * 128 + t * 16 + m;
            const float bv = b1[c];
#pragma unroll
            for (int j = 0; j < 8; ++j) {
                float v = acc[t][j] + bv;
                hbase[(j + 8 * h) * HP + c] = v > 0.f ? v : 0.f;
            }
        }
    }
    __syncthreads();

    // ---- layer 2: out[16,128] = hidden @ W2 + b2 ----
    {
        const float* hrow = hbase + m * HP + 2 * h;
        const float* bp2  = W2p + h * 256 + 2 * m;              // 2N = 256
        v8f acc[8] = {};
#pragma unroll 4
        for (int s = 0; s < 64; ++s) {                          // K = 256
            v2f a = *(const v2f*)(hrow + 4 * s);
            const float* bs = bp2 + (size_t)s * 512;
#pragma unroll
            for (int t = 0; t < 8; ++t) {
                v2f b = *(const v2f*)(bs + 32 * t);
                acc[t] = wmma4(a, b, acc[t]);
            }
        }
#pragma unroll
        for (int t = 0; t < 8; ++t) {
            const int c = t * 16 + m;
            const float bv = b2[c];
#pragma unroll
            for (int j = 0; j < 8; ++j) {
                int r = row0 + j + 8 * h;
                if (r < M) Out[(size_t)r * 128 + c] = acc[t][j] + bv;
            }
        }
    }
}

// ---------------------------------------------------------------------------
extern "C" void kernel_launch(void* const* d_in, const int* in_sizes, int n_in,
                              void* d_out, int out_size, void* d_ws, size_t ws_size,
                              hipStream_t stream) {
    const float* x_user = (const float*)d_in[0];
    const float* x_tx   = (const float*)d_in[1];
    const int*   e_ut   = (const int*)d_in[2];
    const int*   e_tu   = (const int*)d_in[3];
    const float* W_au   = (const float*)d_in[4];
    const float* b_au   = (const float*)d_in[5];
    const float* W_at   = (const float*)d_in[6];
    const float* b_at   = (const float*)d_in[7];
    const float* eps_ut = (const float*)d_in[8];
    const float* eps_tu = (const float*)d_in[9];
    const float* W1_ut  = (const float*)d_in[10];
    const float* b1_ut  = (const float*)d_in[11];
    const float* W2_ut  = (const float*)d_in[12];
    const float* b2_ut  = (const float*)d_in[13];
    const float* W1_tu  = (const float*)d_in[14];
    const float* b1_tu  = (const float*)d_in[15];
    const float* W2_tu  = (const float*)d_in[16];
    const float* b2_tu  = (const float*)d_in[17];

    const int N_USER = 50000, N_TX = 50000, E = 600000;

    float* out_user = (float*)d_out;                    // tuple order: user, tx
    float* out_tx   = out_user + (size_t)N_USER * 128;

    // workspace layout
    float* a_buf  = (float*)d_ws;                       // 6.4M floats
    float* agg    = a_buf  + (size_t)50000 * 128;       // 6.4M floats
    float* wp_au  = agg    + (size_t)50000 * 128;       // 128*128
    float* wp_at  = wp_au  + 128 * 128;                 // 64*128
    float* wp1_ut = wp_at  + 64 * 128;                  // 192*256
    float* wp2_ut = wp1_ut + 192 * 256;                 // 256*128
    float* wp1_tu = wp2_ut + 256 * 128;                 // 256*256
    float* wp2_tu = wp1_tu + 256 * 256;                 // 256*128

    const dim3 bAlign(128), bMlp(64), bFlat(256);
    const int gAlign = (50000 + 63) / 64;               // 782
    const int gMlp   = (50000 + 31) / 32;               // 1563
    const int n4     = 50000 * 128 / 4;
    const int gZero  = (n4 + 255) / 256;
    const int gScat  = (E * 32 + 255) / 256;
    auto gp = [](int kn) { return (kn + 255) / 256; };

    // pack all weights into pair layout (tiny, runs every call)
    pack_kernel<<<gp(128 * 128), bFlat, 0, stream>>>(W_au,  wp_au,  128, 128);
    pack_kernel<<<gp(64  * 128), bFlat, 0, stream>>>(W_at,  wp_at,  64,  128);
    pack_kernel<<<gp(192 * 256), bFlat, 0, stream>>>(W1_ut, wp1_ut, 192, 256);
    pack_kernel<<<gp(256 * 128), bFlat, 0, stream>>>(W2_ut, wp2_ut, 256, 128);
    pack_kernel<<<gp(256 * 256), bFlat, 0, stream>>>(W1_tu, wp1_tu, 256, 256);
    pack_kernel<<<gp(256 * 128), bFlat, 0, stream>>>(W2_tu, wp2_tu, 256, 128);

    // relation (user -> tx): updates tx nodes
    align_kernel<128><<<gAlign, bAlign, 0, stream>>>(x_user, wp_au, b_au, a_buf, N_USER);
    zero_kernel<<<gZero, bFlat, 0, stream>>>((float4*)agg, n4);
    scatter_kernel<<<gScat, bFlat, 0, stream>>>(a_buf, e_ut, E, agg);
    mlp_kernel<64><<<gMlp, bMlp, 0, stream>>>(x_tx, agg, eps_ut,
                                              wp1_ut, b1_ut, wp2_ut, b2_ut,
                                              out_tx, N_TX);

    // relation (tx -> user): updates user nodes
    align_kernel<64><<<gAlign, bAlign, 0, stream>>>(x_tx, wp_at, b_at, a_buf, N_TX);
    zero_kernel<<<gZero, bFlat, 0, stream>>>((float4*)agg, n4);
    scatter_kernel<<<gScat, bFlat, 0, stream>>>(a_buf, e_tu, E, agg);
    mlp_kernel<128><<<gMlp, bMlp, 0, stream>>>(x_user, agg, eps_tu,
                                               wp1_tu, b1_tu, wp2_tu, b2_tu,
                                               out_user, N_USER);
}